// RQVAEModel_30648886624710
// MI455X (gfx1250) — compile-verified
//
#include <hip/hip_runtime.h>
#include <math.h>

#define N_LAT   262144
#define KCB     256
#define DD      64
#define NSTAGE  4
#define SK_IT   20
#define BETA    0.25
#define EPS     0.003f
#define XSTR    68          // LDS row stride (floats): 4n mod 64 bank spread, float2-aligned
#define TROWS   128         // rows per block in dist kernel
#define RPW     32          // rows per wave in sinkhorn row pass
#define SINKB   (N_LAT / (8 * RPW))   // 1024 row-pass blocks

typedef __attribute__((ext_vector_type(2))) float v2f;
typedef __attribute__((ext_vector_type(8))) float v8f;

__device__ __forceinline__ unsigned enc_f32(float f) {
  unsigned u = __float_as_uint(f);
  return u ^ ((unsigned)((int)u >> 31) | 0x80000000u);
}
__device__ __forceinline__ float dec_f32(unsigned u) {
  unsigned mask = (u & 0x80000000u) ? 0x80000000u : 0xFFFFFFFFu;
  return __uint_as_float(u ^ mask);
}

// ---------------- init: residual = x, out_xq = 0 ----------------
__global__ void k_init(const float* __restrict__ x, float* __restrict__ res,
                       float* __restrict__ out_xq) {
  size_t i = (size_t)blockIdx.x * blockDim.x + threadIdx.x;
  if (i < (size_t)N_LAT * DD) { res[i] = x[i]; out_xq[i] = 0.0f; }
}

// ---------------- per-stage prep: reset logv + min/max sentinels ----------------
__global__ void k_prep0(unsigned* __restrict__ minmax, float* __restrict__ logv) {
  if (threadIdx.x < KCB) logv[threadIdx.x] = 0.0f;
  if (threadIdx.x == 0) { minmax[0] = 0xFFFFFFFFu; minmax[1] = 0u; }
}

// ---------------- distance GEMM via V_WMMA_F32_16X16X4_F32 ----------------
// d[i][j] = |x_i|^2 + |c_j|^2 - 2 x_i . c_j  ;  also global min/max (encoded-uint atomics)
__global__ __launch_bounds__(256) void k_dist(const float* __restrict__ res,
                                              const float* __restrict__ cb,
                                              float* __restrict__ dmat,
                                              unsigned* __restrict__ minmax) {
  extern __shared__ float sm[];
  float* cb_s = sm;                       // [KCB][XSTR]
  float* x_s  = cb_s + KCB * XSTR;        // [TROWS][XSTR]
  float* cn_s = x_s + TROWS * XSTR;       // [KCB]
  float* xn_s = cn_s + KCB;               // [TROWS]
  const int tid = threadIdx.x;
  const size_t row_base = (size_t)blockIdx.x * TROWS;

  for (int i = tid; i < KCB * DD; i += 256)
    cb_s[(i >> 6) * XSTR + (i & 63)] = cb[i];
  for (int i = tid; i < TROWS * DD; i += 256)
    x_s[(i >> 6) * XSTR + (i & 63)] = res[(row_base + (size_t)(i >> 6)) * DD + (i & 63)];
  __syncthreads();

  if (tid < KCB) {
    float s = 0.0f;
    #pragma unroll
    for (int k = 0; k < DD; ++k) { float v = cb_s[tid * XSTR + k]; s += v * v; }
    cn_s[tid] = s;
  }
  if (tid < TROWS) {
    float s = 0.0f;
    #pragma unroll
    for (int k = 0; k < DD; ++k) { float v = x_s[tid * XSTR + k]; s += v * v; }
    xn_s[tid] = s;
  }
  __syncthreads();

  const int wave = tid >> 5, lane = tid & 31;
  const int lhalf = lane >> 4, l16 = lane & 15;
  const int koff = lhalf * 2;               // A/B layout: lanes 0-15 -> K 0,1 ; lanes 16-31 -> K 2,3
  const int arow = wave * 16 + l16;         // A-matrix row this lane supplies
  float lmin = 3.4e38f, lmax = -3.4e38f;

  for (int ct = 0; ct < 16; ++ct) {         // 16 column tiles of 16
    v8f acc = {};
    #pragma unroll
    for (int kk = 0; kk < 16; ++kk) {       // K = 64 in steps of 4
      int k = kk * 4 + koff;
      v2f a = *(const v2f*)&x_s[arow * XSTR + k];
      v2f b = *(const v2f*)&cb_s[(ct * 16 + l16) * XSTR + k];
      acc = __builtin_amdgcn_wmma_f32_16x16x4_f32(false, a, false, b,
                                                  (short)0, acc, false, false);
    }
    #pragma unroll
    for (int j = 0; j < 8; ++j) {           // C/D layout: M = 8*half + j, N = lane%16
      int mloc = wave * 16 + lhalf * 8 + j;
      int n = ct * 16 + l16;
      float d = xn_s[mloc] + cn_s[n] - 2.0f * acc[j];
      dmat[(row_base + (size_t)mloc) * KCB + n] = d;
      lmin = fminf(lmin, d); lmax = fmaxf(lmax, d);
    }
  }
  #pragma unroll
  for (int off = 16; off; off >>= 1) {
    lmin = fminf(lmin, __shfl_xor(lmin, off, 32));
    lmax = fmaxf(lmax, __shfl_xor(lmax, off, 32));
  }
  if (lane == 0) {
    atomicMin(&minmax[0], enc_f32(lmin));
    atomicMax(&minmax[1], enc_f32(lmax));
  }
}

// ---------------- center/scale params ----------------
__global__ void k_prep1(const unsigned* __restrict__ minmax, float* __restrict__ params) {
  float mn = dec_f32(minmax[0]), mx = dec_f32(minmax[1]);
  float middle = 0.5f * (mx + mn);
  float amp = mx - middle + 1e-5f;
  params[0] = middle;
  params[1] = -1.0f / (amp * EPS);   // logK = (d - middle) * params[1]
}

// ---------------- one Sinkhorn sweep: fresh log-u per row, per-block column partials ----------------
// cpart layout is TRANSPOSED: cpart[col * SINKB + block]  (coalesced column reduction)
__global__ __launch_bounds__(256) void k_sinkrow(const float* __restrict__ dmat,
                                                 const float* __restrict__ params,
                                                 const float* __restrict__ logv,
                                                 double* __restrict__ cpart) {
  __shared__ double cs[8 * KCB];
  __shared__ float lv[KCB];
  const int tid = threadIdx.x, wave = tid >> 5, lane = tid & 31;
  for (int i = tid; i < 8 * KCB; i += 256) cs[i] = 0.0;
  if (tid < KCB) lv[tid] = logv[tid];
  __syncthreads();

  const float middle = params[0], ninv = params[1];
  const float logB = logf((float)N_LAT);
  double* mycs = cs + wave * KCB;
  const size_t row0 = ((size_t)blockIdx.x * 8 + wave) * RPW;

  for (int r = 0; r < RPW; ++r) {
    const float* drow = dmat + (row0 + r) * KCB;
    float av[2][4];
    float m = -3.4e38f;
    #pragma unroll
    for (int t = 0; t < 2; ++t) {
      float4 dv = *(const float4*)&drow[t * 128 + lane * 4];
      float dk[4] = {dv.x, dv.y, dv.z, dv.w};
      #pragma unroll
      for (int c = 0; c < 4; ++c) {
        int col = t * 128 + lane * 4 + c;
        float a = (dk[c] - middle) * ninv + lv[col];   // logK + logv
        av[t][c] = a;
        m = fmaxf(m, a);
      }
    }
    #pragma unroll
    for (int off = 16; off; off >>= 1) m = fmaxf(m, __shfl_xor(m, off, 32));
    float s = 0.0f;
    #pragma unroll
    for (int t = 0; t < 2; ++t)
      #pragma unroll
      for (int c = 0; c < 4; ++c) s += __expf(av[t][c] - m);
    #pragma unroll
    for (int off = 16; off; off >>= 1) s += __shfl_xor(s, off, 32);
    float logu = -(logB + m + logf(s));                // row normalize (B = N)
    #pragma unroll
    for (int t = 0; t < 2; ++t)
      #pragma unroll
      for (int c = 0; c < 4; ++c) {
        int col = t * 128 + lane * 4 + c;
        // exp(logK+logu+logv) = Q_ij <= 1/B : no overflow in f32
        mycs[col] += (double)__expf(av[t][c] + logu);
      }
  }
  __syncthreads();
  for (int i = tid; i < KCB; i += 256) {
    double s = 0.0;
    #pragma unroll
    for (int w = 0; w < 8; ++w) s += cs[w * KCB + i];
    cpart[(size_t)i * SINKB + blockIdx.x] = s;         // deterministic: no atomics
  }
}

// ---------------- column update: one block per column, coalesced partial reduce ----------------
__global__ __launch_bounds__(256) void k_sinkcol(const double* __restrict__ cpart,
                                                 float* __restrict__ logv) {
  __shared__ double smd[256];
  const int j = blockIdx.x;
  double s = 0.0;
  for (int b = threadIdx.x; b < SINKB; b += 256) s += cpart[(size_t)j * SINKB + b];
  smd[threadIdx.x] = s;
  __syncthreads();
  for (int off = 128; off; off >>= 1) {
    if (threadIdx.x < off) smd[threadIdx.x] += smd[threadIdx.x + off];
    __syncthreads();
  }
  if (threadIdx.x == 0) {
    double t = smd[0];
    if (t < 1e-300) t = 1e-300;
    logv[j] -= (float)log((double)KCB * t);            // v <- v / (K * colsum)
  }
}

// ---------------- argmax + gather + residual/xq/loss update ----------------
__global__ __launch_bounds__(256) void k_assign(const float* __restrict__ dmat,
                                                const float* __restrict__ params,
                                                const float* __restrict__ logv,
                                                const float* __restrict__ cb,
                                                float* __restrict__ res,
                                                float* __restrict__ out_xq,
                                                float* __restrict__ out_idx,
                                                double* __restrict__ lpart,
                                                int stage) {
  __shared__ float lv[KCB];
  __shared__ double ssd_s[8];
  const int tid = threadIdx.x, wave = tid >> 5, lane = tid & 31;
  if (tid < KCB) lv[tid] = logv[tid];
  __syncthreads();
  const float middle = params[0], ninv = params[1];
  const size_t row = (size_t)blockIdx.x * 8 + wave;
  const float* drow = dmat + row * KCB;

  float best = -3.4e38f; int bestj = KCB;
  #pragma unroll
  for (int t = 0; t < 2; ++t) {
    float4 dv = *(const float4*)&drow[t * 128 + lane * 4];
    float dk[4] = {dv.x, dv.y, dv.z, dv.w};
    #pragma unroll
    for (int c = 0; c < 4; ++c) {
      int col = t * 128 + lane * 4 + c;
      float v = (dk[c] - middle) * ninv + lv[col];     // argmax_j Q = argmax_j (logK+logv)
      if (v > best || (v == best && col < bestj)) { best = v; bestj = col; }
    }
  }
  #pragma unroll
  for (int off = 16; off; off >>= 1) {
    float ov = __shfl_xor(best, off, 32);
    int oj = __shfl_xor(bestj, off, 32);
    if (ov > best || (ov == best && oj < bestj)) { best = ov; bestj = oj; }
  }

  const float* crow = cb + (size_t)bestj * DD;
  float ssd = 0.0f;
  #pragma unroll
  for (int t = 0; t < 2; ++t) {
    int c = t * 32 + lane;
    float xq = crow[c];
    float r = res[row * DD + c];
    float df = xq - r;
    ssd += df * df;
    res[row * DD + c] = r - xq;       // residual -= x_q (x_q_st == x_q in value)
    out_xq[row * DD + c] += xq;       // x_q accumulation
  }
  #pragma unroll
  for (int off = 16; off; off >>= 1) ssd += __shfl_xor(ssd, off, 32);
  if (lane == 0) {
    ssd_s[wave] = (double)ssd;
    out_idx[row * NSTAGE + stage] = (float)bestj;
  }
  __syncthreads();
  if (tid == 0) {
    double t = 0.0;
    #pragma unroll
    for (int i = 0; i < 8; ++i) t += ssd_s[i];
    lpart[blockIdx.x] = t;
  }
}

__global__ void k_lossred(const double* __restrict__ lpart, double* __restrict__ lossacc,
                          int stage, int n) {
  __shared__ double smd[256];
  double s = 0.0;
  for (int i = threadIdx.x; i < n; i += 256) s += lpart[i];
  smd[threadIdx.x] = s;
  __syncthreads();
  for (int off = 128; off; off >>= 1) {
    if (threadIdx.x < off) smd[threadIdx.x] += smd[threadIdx.x + off];
    __syncthreads();
  }
  if (threadIdx.x == 0) lossacc[stage] = smd[0];
}

__global__ void k_fin(const double* __restrict__ lossacc, float* __restrict__ out_loss) {
  double acc = 0.0;
  for (int s = 0; s < NSTAGE; ++s)
    acc += (1.0 + BETA) * lossacc[s] / ((double)N_LAT * DD);  // codebook + beta*commit (same mse)
  out_loss[0] = (float)(acc / NSTAGE);
}

// ---------------- host ----------------
extern "C" void kernel_launch(void* const* d_in, const int* in_sizes, int n_in,
                              void* d_out, int out_size, void* d_ws, size_t ws_size,
                              hipStream_t stream) {
  (void)in_sizes; (void)n_in; (void)out_size; (void)ws_size;
  const float* x   = (const float*)d_in[0];
  const float* cbs = (const float*)d_in[1];        // [4][256][64]
  float* out      = (float*)d_out;
  float* out_xq   = out;                           // [N*64]
  float* out_loss = out + (size_t)N_LAT * DD;      // [1]
  float* out_idx  = out_loss + 1;                  // [N*4] (indices as float)

  char* w = (char*)d_ws;
  float* res    = (float*)w;  w += (size_t)N_LAT * DD * 4;
  float* dmat   = (float*)w;  w += (size_t)N_LAT * KCB * 4;
  double* cpart = (double*)w; w += (size_t)SINKB * KCB * 8;
  double* lpart = (double*)w; w += (size_t)(N_LAT / 8) * 8;
  double* lossacc = (double*)w; w += 64;
  float* logv   = (float*)w;  w += KCB * 4;
  float* params = (float*)w;  w += 64;
  unsigned* minmax = (unsigned*)w; w += 64;

  k_init<<<(N_LAT * DD) / 256, 256, 0, stream>>>(x, res, out_xq);

  const size_t smem = (size_t)(KCB * XSTR + TROWS * XSTR + KCB + TROWS) * 4;
  for (int s = 0; s < NSTAGE; ++s) {
    const float* cb = cbs + (size_t)s * KCB * DD;
    k_prep0<<<1, 256, 0, stream>>>(minmax, logv);
    k_dist<<<N_LAT / TROWS, 256, smem, stream>>>(res, cb, dmat, minmax);
    k_prep1<<<1, 1, 0, stream>>>(minmax, params);
    for (int it = 0; it < SK_IT; ++it) {
      k_sinkrow<<<SINKB, 256, 0, stream>>>(dmat, params, logv, cpart);
      k_sinkcol<<<KCB, 256, 0, stream>>>(cpart, logv);
    }
    k_assign<<<N_LAT / 8, 256, 0, stream>>>(dmat, params, logv, cb, res,
                                            out_xq, out_idx, lpart, s);
    k_lossred<<<1, 256, 0, stream>>>(lpart, lossacc, s, N_LAT / 8);
  }
  k_fin<<<1, 1, 0, stream>>>(lossacc, out_loss);
}